// ConditionalRandomField_79104707657741
// MI455X (gfx1250) — compile-verified
//
#include <hip/hip_runtime.h>

#define Bc 64
#define Tc 4096
#define Kc 48

typedef float v2f __attribute__((ext_vector_type(2)));
typedef float v8f __attribute__((ext_vector_type(8)));

#define LOG2E 1.44269504088896340736f
#define LN2   0.69314718055994530942f

// ---------------------------------------------------------------------------
// Kernel 1: target[b] = point_score[b] + trans_score[b]
// one block per batch element, 8 waves strided over t
// ---------------------------------------------------------------------------
__launch_bounds__(256)
__global__ void crf_target_kernel(const float* __restrict__ y_true,
                                  const float* __restrict__ y_pred,
                                  const float* __restrict__ trans,
                                  float* __restrict__ target)
{
    __shared__ float Tl[Kc * Kc];
    __shared__ float wsum[8];
    __shared__ float wtr[8];

    const int tid  = threadIdx.x;
    const int lane = tid & 31;
    const int wv   = tid >> 5;
    const int b    = blockIdx.x;

    for (int i = tid; i < Kc * Kc; i += 256) Tl[i] = trans[i];
    __syncthreads();

    float p_acc = 0.0f;   // per-lane partial of point score
    float t_acc = 0.0f;   // lane0-only partial of transition score

    for (int t = wv; t < Tc; t += 8) {
        const float* yp = y_pred + ((size_t)b * Tc + t) * Kc;
        const float* yt = y_true + ((size_t)b * Tc + t) * Kc;

        float yp0 = yp[lane];
        float yt0 = yt[lane];
        float yp1 = (lane < 16) ? yp[32 + lane] : 0.0f;
        float yt1 = (lane < 16) ? yt[32 + lane] : 0.0f;

        bool okl = (yp0 > -1000000.0f) && (lane >= 16 || yp1 > -1000000.0f);
        bool m0  = (bool)__all((int)okl);

        unsigned long long b0 = __ballot(yt0 > 0.5f);
        unsigned long long b1 = __ballot((lane < 16) && (yt1 > 0.5f));
        int l0 = b0 ? (__ffsll((long long)b0) - 1)
                    : (b1 ? (32 + __ffsll((long long)b1) - 1) : 0);

        if (m0) p_acc += yt0 * yp0 + yt1 * yp1;

        if (t + 1 < Tc) {
            const float* ypn = yp + Kc;
            const float* ytn = yt + Kc;
            float q0 = ypn[lane];
            float z0 = ytn[lane];
            float q1 = (lane < 16) ? ypn[32 + lane] : 0.0f;
            float z1 = (lane < 16) ? ytn[32 + lane] : 0.0f;

            bool okn = (q0 > -1000000.0f) && (lane >= 16 || q1 > -1000000.0f);
            bool m1  = (bool)__all((int)okn);

            unsigned long long c0 = __ballot(z0 > 0.5f);
            unsigned long long c1 = __ballot((lane < 16) && (z1 > 0.5f));
            int l1 = c0 ? (__ffsll((long long)c0) - 1)
                        : (c1 ? (32 + __ffsll((long long)c1) - 1) : 0);

            if (m0 && m1 && lane == 0) t_acc += Tl[l0 * Kc + l1];
        }
    }

    for (int off = 16; off > 0; off >>= 1)
        p_acc += __shfl_xor(p_acc, off, 32);

    if (lane == 0) { wsum[wv] = p_acc; wtr[wv] = t_acc; }
    __syncthreads();

    if (tid == 0) {
        float s = 0.0f;
        for (int i = 0; i < 8; ++i) s += wsum[i] + wtr[i];
        target[b] = s;
    }
}

// ---------------------------------------------------------------------------
// Kernel 2: forward scan in exp-space using V_WMMA_F32_16X16X4_F32
// 4 blocks x 96 threads; block = 16-batch slab; wave w = output cols 16w..16w+15
// Software-pipelined: x_{t+1} global loads issued in phase C of step t.
// ---------------------------------------------------------------------------
__launch_bounds__(96)
__global__ void crf_forward_kernel(const float* __restrict__ y_pred,
                                   const float* __restrict__ trans,
                                   const float* __restrict__ target,
                                   float* __restrict__ out)
{
    __shared__ float Tbuf[Kc * Kc];                      // expT[i][j]
    __shared__ __align__(16) float Sbuf[16 * Kc];        // current state
    __shared__ __align__(16) float Vbuf[16 * Kc];        // exp(S - rowmax)
    __shared__ __align__(16) float Xbuf[16 * Kc];        // x_t slab
    __shared__ __align__(16) float rowmax[16];
    __shared__ __align__(16) float rowmask[16];
    __shared__ float maxpart[3][16];                     // per-wave rowmax partials
    __shared__ float maskp[16 * 6];                      // per-stager mask partials

    const int tid  = threadIdx.x;
    const int lane = tid & 31;
    const int wave = tid >> 5;                   // 0..2  -> N tile
    const int slab = blockIdx.x;                 // 0..3  -> rows slab*16..+15

    // preload expT
    for (int i = tid; i < Kc * Kc; i += 96)
        Tbuf[i] = __builtin_amdgcn_exp2f(trans[i] * LOG2E);

    // WMMA lane mapping
    const int colInTile = lane & 15;
    const int col   = wave * 16 + colInTile;     // output column j
    const int rbase = (lane >> 4) * 8;           // C rows rbase..rbase+7
    const int m_    = lane & 15;                 // A row
    const int koff  = (lane >> 4) * 2;           // A/B k sub-offset

    // x-staging mapping: 96 threads * 8 floats = 16*48
    const int xr = tid / 6;
    const int xs = tid % 6;
    const int xc = xs * 8;
    const float* xsrc_base = y_pred + ((size_t)(slab * 16 + xr)) * Tc * Kc + xc;

    __syncthreads();  // Tbuf visible

    // Constant B fragments: bfrag[k0/4] = expT rows (k0+koff, k0+koff+1), col
    v2f bfrag[12];
    #pragma unroll
    for (int kk = 0; kk < 12; ++kk) {
        int k0 = kk * 4;
        v2f bf;
        bf.x = Tbuf[(k0 + koff)     * Kc + col];
        bf.y = Tbuf[(k0 + koff + 1) * Kc + col];
        bfrag[kk] = bf;
    }

    // ---- Phase A0: stage x(0) + mask partials ----
    {
        const float* src = xsrc_base;            // t = 0
        float4 a = *(const float4*)(src);
        float4 c = *(const float4*)(src + 4);
        *(float4*)&Xbuf[xr * Kc + xc]     = a;
        *(float4*)&Xbuf[xr * Kc + xc + 4] = c;
        bool ok = (a.x > -1000000.0f) && (a.y > -1000000.0f) &&
                  (a.z > -1000000.0f) && (a.w > -1000000.0f) &&
                  (c.x > -1000000.0f) && (c.y > -1000000.0f) &&
                  (c.z > -1000000.0f) && (c.w > -1000000.0f);
        maskp[xr * 6 + xs] = ok ? 1.0f : 0.0f;
    }
    __syncthreads();
    // ---- Phase B0: combine mask ----
    if (tid < 16) {
        float m = maskp[tid * 6];
        for (int q = 1; q < 6; ++q) m = fminf(m, maskp[tid * 6 + q]);
        rowmask[tid] = m;
    }
    __syncthreads();

    // ---- Phase C0: init state, rowmax partials, prefetch x(1) ----
    float4 pxa = *(const float4*)(xsrc_base + (size_t)1 * Kc);
    float4 pxc = *(const float4*)(xsrc_base + (size_t)1 * Kc + 4);

    float sreg[8];
    {
        float4 mk0 = *(const float4*)&rowmask[rbase];
        float4 mk1 = *(const float4*)&rowmask[rbase + 4];
        float mks[8] = {mk0.x, mk0.y, mk0.z, mk0.w, mk1.x, mk1.y, mk1.z, mk1.w};
        #pragma unroll
        for (int i = 0; i < 8; ++i) {
            int r = rbase + i;
            sreg[i] = mks[i] * Xbuf[r * Kc + col];
            Sbuf[r * Kc + col] = sreg[i];
        }
    }
    {
        float pm[8];
        #pragma unroll
        for (int i = 0; i < 8; ++i) pm[i] = sreg[i];
        #pragma unroll
        for (int off = 1; off < 16; off <<= 1) {
            #pragma unroll
            for (int i = 0; i < 8; ++i)
                pm[i] = fmaxf(pm[i], __shfl_xor(pm[i], off, 32));
        }
        if ((lane & 15) == 0) {
            #pragma unroll
            for (int i = 0; i < 8; ++i) maxpart[wave][rbase + i] = pm[i];
        }
    }
    __syncthreads();

    // ---- main scan ----
    #pragma unroll 1
    for (int t = 1; t < Tc; ++t) {
        // Phase A: store prefetched x_t + mask partials; combine rowmax
        {
            *(float4*)&Xbuf[xr * Kc + xc]     = pxa;
            *(float4*)&Xbuf[xr * Kc + xc + 4] = pxc;
            bool ok = (pxa.x > -1000000.0f) && (pxa.y > -1000000.0f) &&
                      (pxa.z > -1000000.0f) && (pxa.w > -1000000.0f) &&
                      (pxc.x > -1000000.0f) && (pxc.y > -1000000.0f) &&
                      (pxc.z > -1000000.0f) && (pxc.w > -1000000.0f);
            maskp[xr * 6 + xs] = ok ? 1.0f : 0.0f;
        }
        if (tid < 16) {
            rowmax[tid] = fmaxf(fmaxf(maxpart[0][tid], maxpart[1][tid]),
                                maxpart[2][tid]);
        }
        __syncthreads();

        // Phase B: V = exp(S - rowmax); combine mask partials
        {
            float mx = rowmax[xr];
            #pragma unroll
            for (int u = 0; u < 8; ++u)
                Vbuf[xr * Kc + xc + u] =
                    __builtin_amdgcn_exp2f((Sbuf[xr * Kc + xc + u] - mx) * LOG2E);
        }
        if (tid < 16) {
            float m = maskp[tid * 6];
            #pragma unroll
            for (int q = 1; q < 6; ++q) m = fminf(m, maskp[tid * 6 + q]);
            rowmask[tid] = m;
        }
        __syncthreads();

        // Phase C: prefetch x_{t+1} (clamped; overlaps WMMA + update + barrier)
        {
            size_t tn = (size_t)((t + 1 < Tc) ? (t + 1) : (Tc - 1));
            pxa = *(const float4*)(xsrc_base + tn * Kc);
            pxc = *(const float4*)(xsrc_base + tn * Kc + 4);
        }

        // acc = V(16xK) * expT(KxN_tile) via 12 x wmma f32 16x16x4
        v8f acc = {};
        #pragma unroll
        for (int kk = 0; kk < 12; ++kk) {
            int k0 = kk * 4;
            v2f a = *(const v2f*)&Vbuf[m_ * Kc + k0 + koff];  // ds_load_b64
            acc = __builtin_amdgcn_wmma_f32_16x16x4_f32(
                false, a, false, bfrag[kk], (short)0, acc, false, false);
        }

        // branchless state update: s = m*new + (1-m)*s  (reference semantics)
        {
            float4 mx0 = *(const float4*)&rowmax[rbase];
            float4 mx1 = *(const float4*)&rowmax[rbase + 4];
            float4 mk0 = *(const float4*)&rowmask[rbase];
            float4 mk1 = *(const float4*)&rowmask[rbase + 4];
            float mxs[8] = {mx0.x, mx0.y, mx0.z, mx0.w, mx1.x, mx1.y, mx1.z, mx1.w};
            float mks[8] = {mk0.x, mk0.y, mk0.z, mk0.w, mk1.x, mk1.y, mk1.z, mk1.w};
            #pragma unroll
            for (int i = 0; i < 8; ++i) {
                int r = rbase + i;
                float ns = __builtin_amdgcn_logf(acc[i]) * LN2
                           + mxs[i] + Xbuf[r * Kc + col];
                sreg[i] = fmaf(mks[i], ns, (1.0f - mks[i]) * sreg[i]);
                Sbuf[r * Kc + col] = sreg[i];
            }
        }

        // new rowmax partials via intra-half-wave shuffle reduction
        {
            float pm[8];
            #pragma unroll
            for (int i = 0; i < 8; ++i) pm[i] = sreg[i];
            #pragma unroll
            for (int off = 1; off < 16; off <<= 1) {
                #pragma unroll
                for (int i = 0; i < 8; ++i)
                    pm[i] = fmaxf(pm[i], __shfl_xor(pm[i], off, 32));
            }
            if ((lane & 15) == 0) {
                #pragma unroll
                for (int i = 0; i < 8; ++i) maxpart[wave][rbase + i] = pm[i];
            }
        }
        __syncthreads();
    }

    // final: out[b] = logsumexp_k(S[b]) - target[b]
    if (tid < 16) {
        const int b = slab * 16 + tid;
        float mx = Sbuf[tid * Kc];
        for (int k = 1; k < Kc; ++k) mx = fmaxf(mx, Sbuf[tid * Kc + k]);
        float s = 0.0f;
        for (int k = 0; k < Kc; ++k)
            s += __builtin_amdgcn_exp2f((Sbuf[tid * Kc + k] - mx) * LOG2E);
        out[b] = __builtin_amdgcn_logf(s) * LN2 + mx - target[b];
    }
}

// ---------------------------------------------------------------------------
extern "C" void kernel_launch(void* const* d_in, const int* in_sizes, int n_in,
                              void* d_out, int out_size, void* d_ws, size_t ws_size,
                              hipStream_t stream) {
    const float* y_true = (const float*)d_in[0];   // (B,T,K) f32 one-hot
    const float* y_pred = (const float*)d_in[1];   // (B,T,K) f32
    const float* trans  = (const float*)d_in[2];   // (K,K)   f32
    float* out    = (float*)d_out;                 // (B,) f32
    float* target = (float*)d_ws;                  // B floats of scratch

    crf_target_kernel<<<Bc, 256, 0, stream>>>(y_true, y_pred, trans, target);
    crf_forward_kernel<<<4, 96, 0, stream>>>(y_pred, trans, target, out);
}